// VanillaClassifierV1_7292854469120
// MI455X (gfx1250) — compile-verified
//
#include <hip/hip_runtime.h>
#include <math.h>
#include <stdint.h>

// Problem shape (fixed by the reference).
#define B_ROWS 8192
#define C_COLS 32000

// Kernel-1 tiling: 320 threads (10 wave32), 8000 float4 per row -> 25 iters/thread exact.
#define TPB        320
#define NWAVES     (TPB / 32)
#define V4_PER_ROW (C_COLS / 4)          // 8000
#define ITERS      (V4_PER_ROW / TPB)    // 25
#define NSTAGES    8                     // LDS ring stages (power of 2)
#define LOOKAHEAD  7                     // async loads in flight per wave
#define STAGE_BYTES (TPB * 16u)          // 5120 B per stage

#define L2E 1.4426950408889634f          // log2(e)
#define LN2 0.6931471805599453f          // ln(2)

// gfx1250 async copy: copies 16 bytes from global sbase plus byte offset voff to LDS address lds_off. Tracked by ASYNCcnt.
__device__ __forceinline__ void async_ld_b128(uint32_t lds_off, uint32_t voff,
                                              const float* sbase) {
  asm volatile("global_load_async_to_lds_b128 %0, %1, %2 offset:0"
               :: "v"(lds_off), "v"(voff), "s"(sbase)
               : "memory");
}

__global__ __launch_bounds__(TPB) void row_nll_kernel(const float* __restrict__ pred,
                                                      const int* __restrict__ target,
                                                      float* __restrict__ row_loss) {
  __shared__ float4 buf[NSTAGES * TPB];
  __shared__ float red_m[NWAVES];
  __shared__ float red_s[NWAVES];

  const int row = blockIdx.x;
  const int tid = threadIdx.x;
  const float* rowp = pred + (size_t)row * C_COLS;   // uniform -> SGPR base for GVS mode

  // Flat address of LDS has the wave-relative byte offset in its low 32 bits.
  const uint32_t lds_base = (uint32_t)(uintptr_t)(void*)buf;
  const uint32_t lane_b   = (uint32_t)tid * 16u;

  // Prologue: fill LOOKAHEAD stages of the ring.
#pragma unroll
  for (int k = 0; k < LOOKAHEAD; ++k) {
    const uint32_t voff = (uint32_t)(k * TPB + tid) * 16u;
    async_ld_b128(lds_base + (uint32_t)k * STAGE_BYTES + lane_b, voff, rowp);
  }

  // Gather the target logit early (thread 0 only) to overlap its latency.
  float xt = 0.0f;
  if (tid == 0) {
    const int tg = target[row];
    xt = rowp[tg];
  }

  // Online softmax accumulators (raw domain max, base-2 exp sum).
  float m = -__builtin_inff();
  float s = 0.0f;

#pragma unroll 1
  for (int i = 0; i < ITERS; ++i) {
    if (i + LOOKAHEAD < ITERS) {
      const int j = i + LOOKAHEAD;
      const uint32_t voff = (uint32_t)(j * TPB + tid) * 16u;
      async_ld_b128(lds_base + (uint32_t)(j & (NSTAGES - 1)) * STAGE_BYTES + lane_b,
                    voff, rowp);
      // issued = i + 1 + LOOKAHEAD; outstanding <= 7  =>  chunk i has landed.
      asm volatile("s_wait_asynccnt 7" ::: "memory");
    } else {
      asm volatile("s_wait_asynccnt 0" ::: "memory");
    }

    const float4 v = buf[(i & (NSTAGES - 1)) * TPB + tid];

    const float cmax = fmaxf(fmaxf(v.x, v.y), fmaxf(v.z, v.w));
    const float nm   = fmaxf(m, cmax);
    const float t    = nm * L2E;
    const float e0 = __builtin_amdgcn_exp2f(__builtin_fmaf(v.x, L2E, -t));
    const float e1 = __builtin_amdgcn_exp2f(__builtin_fmaf(v.y, L2E, -t));
    const float e2 = __builtin_amdgcn_exp2f(__builtin_fmaf(v.z, L2E, -t));
    const float e3 = __builtin_amdgcn_exp2f(__builtin_fmaf(v.w, L2E, -t));
    const float scale = __builtin_amdgcn_exp2f((m - nm) * L2E);  // exp2(-inf)=0 on iter 0
    s = s * scale + ((e0 + e1) + (e2 + e3));
    m = nm;
  }

  // Wave32 reduction of (m, s).
#pragma unroll
  for (int off = 16; off > 0; off >>= 1) {
    const float om = __shfl_xor(m, off, 32);
    const float os = __shfl_xor(s, off, 32);
    const float nm = fmaxf(m, om);
    s = s * __builtin_amdgcn_exp2f((m - nm) * L2E)
      + os * __builtin_amdgcn_exp2f((om - nm) * L2E);
    m = nm;
  }

  const int wid  = tid >> 5;
  const int lane = tid & 31;
  if (lane == 0) { red_m[wid] = m; red_s[wid] = s; }
  __syncthreads();

  if (tid == 0) {
    float M = red_m[0];
    float S = red_s[0];
#pragma unroll
    for (int w = 1; w < NWAVES; ++w) {
      const float om = red_m[w];
      const float os = red_s[w];
      const float nm = fmaxf(M, om);
      S = S * __builtin_amdgcn_exp2f((M - nm) * L2E)
        + os * __builtin_amdgcn_exp2f((om - nm) * L2E);
      M = nm;
    }
    // loss = ln(sum exp(x - M)) + M - x_t = ln2*log2(S) + M - x_t
    const float loss = __builtin_fmaf(LN2, __builtin_amdgcn_logf(S), M) - xt;
    row_loss[row] = loss;
  }
}

__global__ __launch_bounds__(256) void sum_kernel(const float* __restrict__ row_loss,
                                                  float* __restrict__ out) {
  __shared__ float partial[8];
  const int tid = threadIdx.x;

  float acc = 0.0f;
#pragma unroll
  for (int i = tid; i < B_ROWS; i += 256) acc += row_loss[i];

#pragma unroll
  for (int off = 16; off > 0; off >>= 1) acc += __shfl_xor(acc, off, 32);

  if ((tid & 31) == 0) partial[tid >> 5] = acc;
  __syncthreads();

  if (tid == 0) {
    float t = 0.0f;
#pragma unroll
    for (int w = 0; w < 8; ++w) t += partial[w];
    out[0] = t * (1.0f / (float)B_ROWS);
  }
}

extern "C" void kernel_launch(void* const* d_in, const int* in_sizes, int n_in,
                              void* d_out, int out_size, void* d_ws, size_t ws_size,
                              hipStream_t stream) {
  const float* pred   = (const float*)d_in[0];
  const int*   target = (const int*)d_in[1];   // harness passes integer inputs as int32
  float* rowloss = (float*)d_ws;               // 8192 floats of scratch
  float* out     = (float*)d_out;

  row_nll_kernel<<<B_ROWS, TPB, 0, stream>>>(pred, target, rowloss);
  sum_kernel<<<1, 256, 0, stream>>>(rowloss, out);
}